// TopKDecoder_42949672961257
// MI455X (gfx1250) — compile-verified
//
#include <hip/hip_runtime.h>
#include <math.h>

// Problem constants (from reference): B=16, K=8, V=32000, H=512, T=32
#define BB 16
#define KK 8
#define BK 128          // B*K beams
#define VV 32000
#define HH 512
#define H3 1536         // 3*H
#define TT 32
#define SOS_TOK 1
#define EOS_TOK 2
#define NEG_INF (-__builtin_inff())

#define MROWS 64        // A rows staged in LDS per block (4 M-tiles, 128 KB)

typedef __attribute__((ext_vector_type(2))) float v2f;
typedef __attribute__((ext_vector_type(8))) float v8f;

// ---------------------------------------------------------------------------
// init: hidden0 = tile(encoder_hidden, K)  (hidden0[m] = enc[m % B]),
// tokens = SOS, seq_scores = 0 on beam 0 of each batch else -inf
// ---------------------------------------------------------------------------
__global__ void init_kernel(const float* __restrict__ enc,
                            float* __restrict__ hidden,
                            int* __restrict__ tokens,
                            float* __restrict__ seq_scores) {
    int idx = blockIdx.x * blockDim.x + threadIdx.x;
    if (idx < BK * HH) {
        int m = idx >> 9;          // / HH
        int j = idx & (HH - 1);
        hidden[idx] = enc[(m % BB) * HH + j];
    }
    if (blockIdx.x == 0 && threadIdx.x < BK) {
        tokens[threadIdx.x] = SOS_TOK;
        seq_scores[threadIdx.x] = ((threadIdx.x % KK) == 0) ? 0.f : NEG_INF;
    }
}

// ---------------------------------------------------------------------------
// GRU gate GEMMs via V_WMMA_F32_16X16X4_F32:
//   gi = embedding[tokens] @ w_ih^T + b_ih   (blockIdx.z == 0)
//   gh = hidden            @ w_hh^T + b_hh   (blockIdx.z == 1)
// One wave32 per 16x16 output tile.  A 16x4 layout: lane = M row,
// VGPR pair holds K = {0,1} (lanes 0-15) / {2,3} (lanes 16-31).
// B 4x16 = w^T tile: B[k][n] = w[n*H + k] -> contiguous float2 per lane.
// ---------------------------------------------------------------------------
__global__ __launch_bounds__(32) void gru_gemm_kernel(
    const float* __restrict__ embedding, const float* __restrict__ hidden,
    const int* __restrict__ tokens,
    const float* __restrict__ w_ih, const float* __restrict__ w_hh,
    const float* __restrict__ b_ih, const float* __restrict__ b_hh,
    float* __restrict__ gi, float* __restrict__ gh) {
    const int ntile = blockIdx.x;        // 0..95  (H3/16)
    const int mtile = blockIdx.y;        // 0..7   (BK/16)
    const int which = blockIdx.z;        // 0 = gi, 1 = gh
    const int lane  = threadIdx.x;
    const int l     = lane & 15;
    const int half  = lane >> 4;

    const int m = mtile * 16 + l;        // A row for this lane
    const int n = ntile * 16 + l;        // B/D column for this lane

    const float* arow;
    const float* w;
    const float* bias;
    float* out;
    if (which == 0) {
        arow = embedding + (size_t)tokens[m] * HH;
        w = w_ih; bias = b_ih; out = gi;
    } else {
        arow = hidden + (size_t)m * HH;
        w = w_hh; bias = b_hh; out = gh;
    }
    const float* wrow = w + (size_t)n * HH;

    v8f acc = {};
#pragma unroll 4
    for (int k0 = 0; k0 < HH; k0 += 4) {
        v2f a = *(const v2f*)(arow + k0 + 2 * half);
        v2f b = *(const v2f*)(wrow + k0 + 2 * half);
        acc = __builtin_amdgcn_wmma_f32_16x16x4_f32(false, a, false, b,
                                                    (short)0, acc, false, false);
    }
    const float bn = bias[n];
    const int rbase = mtile * 16 + 8 * half;  // D row = vgpr + 8*half
#pragma unroll
    for (int r = 0; r < 8; ++r)
        out[(size_t)(rbase + r) * H3 + n] = acc[r] + bn;
}

// ---------------------------------------------------------------------------
// Elementwise GRU gates: h_new = (1-z)*n + z*h
// ---------------------------------------------------------------------------
__global__ void gru_gates_kernel(const float* __restrict__ gi,
                                 const float* __restrict__ gh,
                                 const float* __restrict__ hidden,
                                 float* __restrict__ h_new) {
    int idx = blockIdx.x * blockDim.x + threadIdx.x;
    if (idx >= BK * HH) return;
    int m = idx >> 9;
    int j = idx & (HH - 1);
    const float* gim = gi + (size_t)m * H3;
    const float* ghm = gh + (size_t)m * H3;
    float ir = gim[j],        hr = ghm[j];
    float iz = gim[HH + j],   hz = ghm[HH + j];
    float in_ = gim[2*HH + j], hn = ghm[2*HH + j];
    float r = 1.f / (1.f + __expf(-(ir + hr)));
    float z = 1.f / (1.f + __expf(-(iz + hz)));
    float nn = tanhf(in_ + r * hn);
    h_new[idx] = (1.f - z) * nn + z * hidden[idx];
}

// ---------------------------------------------------------------------------
// Logits GEMM: [128,512] @ [512,32000] + b_out, fp32 WMMA.
// Block = 256 threads = 8 wave32.  64 A rows (4 M-tiles, 128 KB) staged in
// LDS; each wave owns one N-column group and reuses every B load across the
// 4 M-tiles (4 accumulator sets) -> 2 strided 4-byte global loads per 4 WMMAs, and
// w_out (L2-resident, 65.5 MB of 192 MB L2) is streamed only 2x per step.
// ---------------------------------------------------------------------------
__global__ __launch_bounds__(256) void logits_kernel(
    const float* __restrict__ h_new, const float* __restrict__ w_out,
    const float* __restrict__ b_out, float* __restrict__ out /*[BK,VV]*/) {
    __shared__ float sA[MROWS * HH];       // 128 KB of 320 KB LDS
    const int mblk = blockIdx.y;           // 0..1 (BK/MROWS)
    const int tid  = threadIdx.x;

    {   // cooperative A stage: contiguous float4 copy (64 rows x 512)
        const float4* src = (const float4*)(h_new + (size_t)mblk * MROWS * HH);
        float4* dst = (float4*)sA;
#pragma unroll
        for (int i = tid; i < (MROWS * HH) / 4; i += 256) dst[i] = src[i];
    }
    __syncthreads();

    const int wave = tid >> 5;
    const int lane = tid & 31;
    const int l    = lane & 15;
    const int half = lane >> 4;
    const int nc   = (blockIdx.x * 8 + wave) * 16 + l;   // output column

    v8f acc0 = {}, acc1 = {}, acc2 = {}, acc3 = {};
    for (int k0 = 0; k0 < HH; k0 += 4) {
        const int k = k0 + 2 * half;
        v2f b;
        b.x = w_out[(size_t)k * VV + nc];
        b.y = w_out[(size_t)(k + 1) * VV + nc];
        if ((k0 & 31) == 0 && k0 + 32 < HH)              // pull next K-slab of B
            __builtin_prefetch(w_out + (size_t)(k0 + 32) * VV + nc, 0, 1);
        const float* sk = sA + l * HH + k;
        v2f a0 = *(const v2f*)(sk);
        v2f a1 = *(const v2f*)(sk + 16 * HH);
        v2f a2 = *(const v2f*)(sk + 32 * HH);
        v2f a3 = *(const v2f*)(sk + 48 * HH);
        acc0 = __builtin_amdgcn_wmma_f32_16x16x4_f32(false, a0, false, b,
                                                     (short)0, acc0, false, false);
        acc1 = __builtin_amdgcn_wmma_f32_16x16x4_f32(false, a1, false, b,
                                                     (short)0, acc1, false, false);
        acc2 = __builtin_amdgcn_wmma_f32_16x16x4_f32(false, a2, false, b,
                                                     (short)0, acc2, false, false);
        acc3 = __builtin_amdgcn_wmma_f32_16x16x4_f32(false, a3, false, b,
                                                     (short)0, acc3, false, false);
    }
    const float bn = b_out[nc];
    const int rbase = mblk * MROWS + 8 * half;
#pragma unroll
    for (int r = 0; r < 8; ++r) {
        out[(size_t)(rbase +  0 + r) * VV + nc] = acc0[r] + bn;
        out[(size_t)(rbase + 16 + r) * VV + nc] = acc1[r] + bn;
        out[(size_t)(rbase + 32 + r) * VV + nc] = acc2[r] + bn;
        out[(size_t)(rbase + 48 + r) * VV + nc] = acc3[r] + bn;
    }
}

// ---------------------------------------------------------------------------
// Row-wise logZ = max + log(sum(exp(x-max))); one block per beam row.
// ---------------------------------------------------------------------------
__global__ __launch_bounds__(256) void logz_kernel(const float* __restrict__ logits,
                                                   float* __restrict__ logZ) {
    __shared__ float sred[256];
    const int m = blockIdx.x;
    const int tid = threadIdx.x;
    const float* row = logits + (size_t)m * VV;
    float mx = NEG_INF;
    for (int v = tid; v < VV; v += 256) mx = fmaxf(mx, row[v]);
    sred[tid] = mx; __syncthreads();
    for (int s = 128; s > 0; s >>= 1) {
        if (tid < s) sred[tid] = fmaxf(sred[tid], sred[tid + s]);
        __syncthreads();
    }
    mx = sred[0]; __syncthreads();
    float sum = 0.f;
    for (int v = tid; v < VV; v += 256) sum += __expf(row[v] - mx);
    sred[tid] = sum; __syncthreads();
    for (int s = 128; s > 0; s >>= 1) {
        if (tid < s) sred[tid] += sred[tid + s];
        __syncthreads();
    }
    if (tid == 0) logZ[m] = mx + __logf(sred[0]);
}

// in-place: logp = logits - logZ[row]
__global__ void subz_kernel(float* __restrict__ out, const float* __restrict__ logZ) {
    size_t idx = (size_t)blockIdx.x * blockDim.x + threadIdx.x;
    if (idx >= (size_t)BK * VV) return;
    out[idx] -= logZ[idx / VV];
}

// ---------------------------------------------------------------------------
// Per-batch top-8 over K*V candidates: register-local top8 + LDS merge.
// Also performs EOS masking and emits stored scores / syms / preds.
// ---------------------------------------------------------------------------
__global__ __launch_bounds__(256) void topk_kernel(
    const float* __restrict__ logp,      // [BK,VV] log-softmax
    float* __restrict__ seq_scores,      // [BK] in/out
    int* __restrict__ tokens,            // [BK] out
    int* __restrict__ preds_l,           // [BK] out (global beam index)
    float* __restrict__ d_scores,        // [BK] stored scores (pre-mask)
    float* __restrict__ d_syms,
    float* __restrict__ d_preds) {
    __shared__ float svals[2048];
    __shared__ int   sidx[2048];
    __shared__ float rv[256];
    __shared__ int   rp[256];
    __shared__ float out_val[KK];
    __shared__ int   out_idx[KK];

    const int b = blockIdx.x;
    const int tid = threadIdx.x;

    float lv[8]; int li[8];
#pragma unroll
    for (int i = 0; i < 8; ++i) { lv[i] = NEG_INF; li[i] = 0; }

    for (int k = 0; k < KK; ++k) {
        const float ss = seq_scores[b * KK + k];
        const float* row = logp + (size_t)(b * KK + k) * VV;
        for (int v = tid; v < VV; v += 256) {
            float c = ss + row[v];
            if (c > lv[7]) {
                int ci = k * VV + v;
                int p = 7;
                while (p > 0 && lv[p - 1] < c) { lv[p] = lv[p - 1]; li[p] = li[p - 1]; --p; }
                lv[p] = c; li[p] = ci;
            }
        }
    }
#pragma unroll
    for (int i = 0; i < 8; ++i) { svals[tid * 8 + i] = lv[i]; sidx[tid * 8 + i] = li[i]; }
    __syncthreads();

    for (int round = 0; round < KK; ++round) {
        float best = NEG_INF; int bpos = 0;
#pragma unroll
        for (int i = 0; i < 8; ++i) {
            float x = svals[tid * 8 + i];
            if (x > best) { best = x; bpos = tid * 8 + i; }
        }
        rv[tid] = best; rp[tid] = bpos; __syncthreads();
        for (int s = 128; s > 0; s >>= 1) {
            if (tid < s && rv[tid + s] > rv[tid]) { rv[tid] = rv[tid + s]; rp[tid] = rp[tid + s]; }
            __syncthreads();
        }
        if (tid == 0) {
            out_val[round] = rv[0];
            out_idx[round] = sidx[rp[0]];
            svals[rp[0]] = NEG_INF;
        }
        __syncthreads();
    }

    if (tid < KK) {
        float sc = out_val[tid];
        int ci = out_idx[tid];
        int tok = ci % VV;
        int pred = ci / VV + b * KK;
        int g = b * KK + tid;
        d_scores[g] = sc;                 // stored before EOS masking
        d_syms[g]   = (float)tok;
        d_preds[g]  = (float)pred;
        tokens[g]   = tok;
        preds_l[g]  = pred;
        seq_scores[g] = (tok == EOS_TOK) ? NEG_INF : sc;
    }
}

// beam reorder: hidden[m] = h_new[preds_l[m]]
__global__ void reorder_kernel(const float* __restrict__ h_new,
                               const int* __restrict__ preds_l,
                               float* __restrict__ hidden) {
    const int m = blockIdx.x;
    const int src = preds_l[m];
    const float* s = h_new + (size_t)src * HH;
    float* d = hidden + (size_t)m * HH;
    for (int j = threadIdx.x; j < HH; j += blockDim.x) d[j] = s[j];
}

__global__ void copyh_kernel(const float* __restrict__ hidden, float* __restrict__ dst) {
    int idx = blockIdx.x * blockDim.x + threadIdx.x;
    if (idx < BK * HH) dst[idx] = hidden[idx];
}

// ---------------------------------------------------------------------------
extern "C" void kernel_launch(void* const* d_in, const int* in_sizes, int n_in,
                              void* d_out, int out_size, void* d_ws, size_t ws_size,
                              hipStream_t stream) {
    (void)in_sizes; (void)n_in; (void)out_size; (void)ws_size;
    const float* enc   = (const float*)d_in[0];
    const float* emb   = (const float*)d_in[1];
    const float* w_ih  = (const float*)d_in[2];
    const float* w_hh  = (const float*)d_in[3];
    const float* b_ih  = (const float*)d_in[4];
    const float* b_hh  = (const float*)d_in[5];
    const float* w_out = (const float*)d_in[6];
    const float* b_out = (const float*)d_in[7];

    float* out = (float*)d_out;
    float* outs = out;                                   // [T,BK,V]
    const size_t off = (size_t)TT * BK * VV;
    float* d_scores = out + off;                         // [T,BK]
    float* d_syms   = d_scores + (size_t)TT * BK;        // [T,BK]
    float* d_preds  = d_syms + (size_t)TT * BK;          // [T,BK]
    float* d_hidf   = d_preds + (size_t)TT * BK;         // [1,BK,H]

    float* ws = (float*)d_ws;
    float* hidden = ws;        ws += BK * HH;
    float* h_new  = ws;        ws += BK * HH;
    float* gi     = ws;        ws += BK * H3;
    float* gh     = ws;        ws += BK * H3;
    float* seqsc  = ws;        ws += BK;
    float* logZ   = ws;        ws += BK;
    int*   tokens = (int*)ws;  ws += BK;
    int*   predsl = (int*)ws;  ws += BK;

    init_kernel<<<256, 256, 0, stream>>>(enc, hidden, tokens, seqsc);

    for (int t = 0; t < TT; ++t) {
        float* logp_t = outs + (size_t)t * BK * VV;
        gru_gemm_kernel<<<dim3(H3 / 16, BK / 16, 2), 32, 0, stream>>>(
            emb, hidden, tokens, w_ih, w_hh, b_ih, b_hh, gi, gh);
        gru_gates_kernel<<<(BK * HH) / 256, 256, 0, stream>>>(gi, gh, hidden, h_new);
        logits_kernel<<<dim3(VV / 16 / 8, BK / MROWS), 256, 0, stream>>>(
            h_new, w_out, b_out, logp_t);
        logz_kernel<<<BK, 256, 0, stream>>>(logp_t, logZ);
        subz_kernel<<<(BK * VV) / 256, 256, 0, stream>>>(logp_t, logZ);
        topk_kernel<<<BB, 256, 0, stream>>>(logp_t, seqsc, tokens, predsl,
                                            d_scores + (size_t)t * BK,
                                            d_syms + (size_t)t * BK,
                                            d_preds + (size_t)t * BK);
        reorder_kernel<<<BK, 256, 0, stream>>>(h_new, predsl, hidden);
    }
    copyh_kernel<<<256, 256, 0, stream>>>(hidden, d_hidf);
}